// ContourIntegrationLayer_22007412424936
// MI455X (gfx1250) — compile-verified
//
#include <hip/hip_runtime.h>
#include <stdint.h>

typedef float v4f __attribute__((ext_vector_type(4)));

#define NCH    96
#define HW     112
#define PLANE  (HW * HW)      // 12544 floats per channel plane
#define PLANE4 (PLANE / 4)    // 3136 float4 per plane

// ---------------------------------------------------------------------------
// Kernel 1: the 92 zero-kernel channels.  out = x + bias[c].
// Pure streaming -> float4 (B128) non-temporal loads/stores.
// Special channels (5,10,54,67) are skipped entirely (kernel 2 owns them).
// ---------------------------------------------------------------------------
__global__ void contour_pass(const float* __restrict__ x,
                             const float* __restrict__ bias,
                             float* __restrict__ out,
                             unsigned total4)
{
    unsigned i = blockIdx.x * blockDim.x + threadIdx.x;
    if (i >= total4) return;
    unsigned c = (i / PLANE4) % NCH;
    if (c == 5u || c == 10u || c == 54u || c == 67u) return;

    v4f xv = __builtin_nontemporal_load(((const v4f*)x) + i);
    v4f o  = xv + bias[c];
    __builtin_nontemporal_store(o, ((v4f*)out) + i);
}

// ---------------------------------------------------------------------------
// Kernel 2: the 4 masked-stencil channels (8 taps each).
// One block = one (batch, special channel, row).  The 8 source rows of the
// stencil are staged into LDS with the CDNA5 async copy path
// (global_load_async_to_lds_b128, ASYNCcnt), then 112 lanes compute
// out = y*x + bias + x from LDS.
//   type 0 = horizontal (ch 5):  taps at (r, c+off)
//   type 1 = vertical   (ch 10): taps at (r+off, c)
//   type 2 = diagonal   (ch 54, 67): taps at (r+off, c+off)
// off = {-12,-9,-6,-3,+3,+6,+9,+12}
// ---------------------------------------------------------------------------
__global__ void contour_special(const float* __restrict__ x,
                                const float* __restrict__ rk,   // raw_kernel (96,25,25)
                                const float* __restrict__ bias,
                                float* __restrict__ out)
{
    __shared__ __align__(16) float tile[8 * HW];   // up to 8 staged rows

    const int tid = threadIdx.x;
    int b = blockIdx.x;
    const int r = b % HW;  b /= HW;
    const int s = b & 3;   b >>= 2;
    const int n = b;

    const int chs[4] = {5, 10, 54, 67};
    const int typ[4] = {0, 1, 2, 2};
    const int KI[8]  = {0, 3, 6, 9, 15, 18, 21, 24};   // kernel index; offset = KI - 12

    const int ch   = chs[s];
    const int type = typ[s];

    const float* xp = x   + (size_t)(n * NCH + ch) * PLANE;
    float*       op = out + (size_t)(n * NCH + ch) * PLANE;

    // --- zero the tile (rows that fall outside the image must read as 0) ---
    const int nrows = (type == 0) ? 1 : 8;
    for (int e = tid; e < nrows * HW; e += blockDim.x) tile[e] = 0.0f;
    __syncthreads();

    // --- stage source rows: async global -> LDS, 16B per lane-transfer ---
    const int nq = nrows * 28;                 // 28 float4 per 112-float row
    for (int e = tid; e < nq; e += blockDim.x) {
        const int rs = e / 28;
        const int q  = e - rs * 28;
        const int sr = (type == 0) ? r : (r + KI[rs] - 12);
        if (sr >= 0 && sr < HW) {
            const float* g = xp + sr * HW + q * 4;
            unsigned lo = (unsigned)(uintptr_t)(&tile[rs * HW + q * 4]);
            asm volatile("global_load_async_to_lds_b128 %0, %1, off"
                         :: "v"(lo), "v"((unsigned long long)(uintptr_t)g)
                         : "memory");
        }
    }
    asm volatile("s_wait_asynccnt 0x0" ::: "memory");
    __syncthreads();

    // --- compute: 8-tap dot from LDS, gate by x, add bias + residual ---
    if (tid < HW) {
        const int c = tid;

        float w[8];
        const float* rkc = rk + ch * 625;
        if (type == 0) {
#pragma unroll
            for (int t = 0; t < 8; ++t) w[t] = rkc[12 * 25 + KI[t]];
        } else if (type == 1) {
#pragma unroll
            for (int t = 0; t < 8; ++t) w[t] = rkc[KI[t] * 25 + 12];
        } else {
#pragma unroll
            for (int t = 0; t < 8; ++t) w[t] = rkc[KI[t] * 26];      // i*25 + i
        }

        float y = 0.0f;
        if (type == 1) {                 // vertical: column c of each staged row
#pragma unroll
            for (int t = 0; t < 8; ++t) y = fmaf(w[t], tile[t * HW + c], y);
        } else if (type == 0) {          // horizontal: shifted columns of row r
#pragma unroll
            for (int t = 0; t < 8; ++t) {
                const int cc = c + KI[t] - 12;
                if (cc >= 0 && cc < HW) y = fmaf(w[t], tile[cc], y);
            }
        } else {                         // diagonal: shifted columns per row
#pragma unroll
            for (int t = 0; t < 8; ++t) {
                const int cc = c + KI[t] - 12;
                if (cc >= 0 && cc < HW) y = fmaf(w[t], tile[t * HW + cc], y);
            }
        }

        const float xv = xp[r * HW + c];
        op[r * HW + c] = fmaf(y, xv, bias[ch] + xv);
    }
}

// ---------------------------------------------------------------------------
extern "C" void kernel_launch(void* const* d_in, const int* in_sizes, int n_in,
                              void* d_out, int out_size, void* d_ws, size_t ws_size,
                              hipStream_t stream)
{
    const float* x    = (const float*)d_in[0];   // (64, 96, 112, 112)
    const float* rk   = (const float*)d_in[1];   // (96, 25, 25)
    const float* bias = (const float*)d_in[2];   // (96,)
    float*       out  = (float*)d_out;

    const unsigned total4 = (unsigned)(out_size / 4);
    const int nblk1 = (int)((total4 + 255u) / 256u);
    contour_pass<<<nblk1, 256, 0, stream>>>(x, bias, out, total4);

    const int nbatch = out_size / (NCH * PLANE);          // 64
    const int nblk2  = nbatch * 4 * HW;                   // batch x special-ch x row
    contour_special<<<nblk2, 128, 0, stream>>>(x, rk, bias, out);
}